// CTCLayer_73160472920694
// MI455X (gfx1250) — compile-verified
//
#include <hip/hip_runtime.h>

// ---------------- problem constants (match reference) ----------------
#define CTC_B   64
#define CTC_T   2048
#define CTC_C   128
#define CTC_L   256
#define CTC_S   513               // 2*L + 1
#define BLANKC  (CTC_C - 1)
#define NEGF    (-1e30f)
#define EPSF    (1e-7f)

#define TILE    16                // frames staged per LDS tile (8 KB)
#define NTHR    512               // 16 wave32 per block
#define NTILES  (CTC_T / TILE)

// ---------------- CDNA5 async global->LDS path (probe via __has_builtin) ----
#define USE_ASYNC 0
#if defined(__HIP_DEVICE_COMPILE__)
#if __has_builtin(__builtin_amdgcn_global_load_async_to_lds_b128) && \
    __has_builtin(__builtin_amdgcn_s_wait_asynccnt)
#undef USE_ASYNC
#define USE_ASYNC 1
#endif
#endif

typedef int v4i_t __attribute__((ext_vector_type(4)));

// Copy one 16-byte chunk global -> LDS (per thread).
__device__ __forceinline__ void stage16(const float* __restrict__ g, float* l) {
#if USE_ASYNC
  __builtin_amdgcn_global_load_async_to_lds_b128(
      (v4i_t*)(const_cast<float*>(g)), (v4i_t*)l, /*offset=*/0, /*cpol=*/0);
#else
  const float4 v = *(const float4*)g;
  *(float4*)l = v;
#endif
}

__device__ __forceinline__ void wait_async(int n) {
#if USE_ASYNC
  if (n == 0) __builtin_amdgcn_s_wait_asynccnt(0);
  else        __builtin_amdgcn_s_wait_asynccnt(1);
#else
  (void)n;
#endif
}

// ---------------------------------------------------------------------------
// One block per batch element. State s = tid (tid 0 also owns s = 512).
// alpha double-buffered in LDS, padded by 2 NEG entries in front so the
// s-1 / s-2 reads are branchless. y_pred staged 16 frames at a time into
// double-buffered LDS tiles via async loads (1 ASYNCcnt group per wave per
// tile; completions are in-order, so s_wait_asynccnt(1) releases exactly
// the next tile while the tile after it stays in flight).
// ---------------------------------------------------------------------------
__global__ __launch_bounds__(NTHR)
void ctc_loss_kernel(const int* __restrict__ y_true,
                     const float* __restrict__ y_pred,
                     float* __restrict__ out) {
  __shared__ float alphaA[CTC_S + 2];
  __shared__ float alphaB[CTC_S + 2];
  __shared__ __align__(16) float tileBuf[2][TILE * CTC_C];

  const int b   = blockIdx.x;
  const int tid = threadIdx.x;
  const float* pred = y_pred + (size_t)b * CTC_T * CTC_C;

  // --- issue tiles 0 and 1 immediately; overlap with label setup below ----
  stage16(pred + tid * 4,                  &tileBuf[0][tid * 4]);
  stage16(pred + TILE * CTC_C + tid * 4,   &tileBuf[1][tid * 4]);

  const int* lab = y_true + b * CTC_L;

  // Per-thread extended-label class and skip flag for state s = tid.
  // ext: blank,l0,blank,l1,...; skip[s] = odd(s) && s>=3 && lab[li]!=lab[li-1].
  int e0, k0;
  if (tid & 1) {
    const int li = tid >> 1;
    e0 = lab[li];
    k0 = (li > 0) && (e0 != lab[li - 1]);
  } else {
    e0 = BLANKC;
    k0 = 0;
  }

  if (tid < 2) { alphaA[tid] = NEGF; alphaB[tid] = NEGF; }

  // tile 0 ready (in-order: waiting to <=1 outstanding completes group 0)
  wait_async(1);
  __syncthreads();

  // --- t = 0 init: alpha0 = NEG except s=0,1 get lp(0,s) -------------------
  {
    const float* row = tileBuf[0];
    const float lp = __logf(row[e0] + EPSF);
    alphaA[2 + tid] = (tid <= 1) ? lp : NEGF;
    if (tid == 0) alphaA[2 + 512] = NEGF;
  }
  __syncthreads();

  float* aOld = alphaA;
  float* aNew = alphaB;

  for (int ti = 0; ti < NTILES; ++ti) {
    const float* rows = tileBuf[ti & 1];
    const int r0 = (ti == 0) ? 1 : 0;

    // speculative L2 prefetch a couple tiles ahead (global_prefetch_b8)
    if (ti + 3 < NTILES && tid < TILE)
      __builtin_prefetch(pred + (size_t)(ti + 3) * TILE * CTC_C +
                         (size_t)tid * CTC_C, 0, 1);

    for (int r = r0; r < TILE; ++r) {
      const float* row = rows + r * CTC_C;
      const float lp = __logf(row[e0] + EPSF);

      const float a0 = aOld[2 + tid];
      const float a1 = aOld[1 + tid];
      const float a2 = k0 ? aOld[tid] : NEGF;
      const float m  = fmaxf(a0, fmaxf(a1, a2));
      const float v  = m + __logf(__expf(a0 - m) + __expf(a1 - m) +
                                  __expf(a2 - m));
      aNew[2 + tid] = v + lp;

      if (tid == 0) {               // extra state s = 512 (blank, no skip)
        const float lpb = __logf(row[BLANKC] + EPSF);
        const float b0 = aOld[2 + 512];
        const float b1 = aOld[1 + 512];
        const float mb = fmaxf(b0, b1);
        aNew[2 + 512] = mb + __logf(__expf(b0 - mb) + __expf(b1 - mb)) + lpb;
      }
      __syncthreads();
      float* t2 = aOld; aOld = aNew; aNew = t2;
    }

    // refill the buffer we just drained with tile ti+2 (stays in flight)
    if (ti + 2 < NTILES)
      stage16(pred + (size_t)(ti + 2) * TILE * CTC_C + tid * 4,
              &tileBuf[ti & 1][tid * 4]);

    // before reading tile ti+1: wait until only the newest group remains
    if (ti + 1 < NTILES) {
      wait_async((ti + 2 < NTILES) ? 1 : 0);
      __syncthreads();
    }
  }

  if (tid == 0) {
    const float x = aOld[2 + CTC_S - 1];
    const float y = aOld[2 + CTC_S - 2];
    const float m = fmaxf(x, y);
    out[b] = -(m + __logf(__expf(x - m) + __expf(y - m)));
  }
}

// ---------------------------------------------------------------------------
extern "C" void kernel_launch(void* const* d_in, const int* in_sizes, int n_in,
                              void* d_out, int out_size, void* d_ws,
                              size_t ws_size, hipStream_t stream) {
  (void)in_sizes; (void)n_in; (void)d_ws; (void)ws_size; (void)out_size;
  const int*   y_true = (const int*)d_in[0];   // [B,L] int32
  const float* y_pred = (const float*)d_in[1]; // [B,T,C] float32
  float*       out    = (float*)d_out;         // [B,1] float32
  ctc_loss_kernel<<<dim3(CTC_B), dim3(NTHR), 0, stream>>>(y_true, y_pred, out);
}